// GNN_22093311771315
// MI455X (gfx1250) — compile-verified
//
#include <hip/hip_runtime.h>
#include <hip/hip_bf16.h>

typedef __attribute__((ext_vector_type(2))) float v2f;
typedef __attribute__((ext_vector_type(8))) float v8f;

#define GNN_N 100000
#define GNN_E 1600000
#define TILES_M 6250   // N / 16

// ---------------------------------------------------------------------------
// Degree accumulation: deg[dst[e]] += 1
// ---------------------------------------------------------------------------
__global__ __launch_bounds__(256) void deg_kernel(const int* __restrict__ dst,
                                                  float* __restrict__ deg) {
    int e = blockIdx.x * 256 + threadIdx.x;
    if (e < GNN_E) {
        unsafeAtomicAdd(&deg[dst[e]], 1.0f);
    }
}

// deg -> deg_inv in place
__global__ __launch_bounds__(256) void deginv_kernel(float* __restrict__ deg) {
    int i = blockIdx.x * 256 + threadIdx.x;
    if (i < GNN_N) {
        float d = deg[i];
        deg[i] = (d > 0.0f) ? (1.0f / d) : 0.0f;
    }
}

// ---------------------------------------------------------------------------
// Edge scatter-add: out[dst[e], :] += feat[src[e], :]
// one thread per (edge, 4-channel chunk); C4 = channels/4
// ---------------------------------------------------------------------------
__global__ __launch_bounds__(256) void scatter_add4(const float* __restrict__ feat,
                                                    const int* __restrict__ src,
                                                    const int* __restrict__ dst,
                                                    float* __restrict__ out,
                                                    int C4, unsigned int total) {
    unsigned int idx = blockIdx.x * 256u + threadIdx.x;
    if (idx >= total) return;
    unsigned int e = idx / (unsigned int)C4;
    unsigned int c = idx - e * (unsigned int)C4;
    int s = src[e];
    int d = dst[e];
    const float4 v = *(const float4*)(feat + ((size_t)s * C4 + c) * 4);
    float* o = out + ((size_t)d * C4 + c) * 4;
    unsafeAtomicAdd(o + 0, v.x);
    unsafeAtomicAdd(o + 1, v.y);
    unsafeAtomicAdd(o + 2, v.z);
    unsafeAtomicAdd(o + 3, v.w);
}

// ---------------------------------------------------------------------------
// Fused SAGE layer GEMM (f32 WMMA 16x16x4):
//   out = BN( (msg * deg_inv) @ Wl + Aself @ Wr + bias ) [+ ReLU]
// One wave computes CT adjacent 16x16 output tiles of one M-tile, sharing a
// single A load across CT WMMAs per K-step (A L2-traffic / CT).
// 8 waves per block.
// ---------------------------------------------------------------------------
template <int CT>
__global__ __launch_bounds__(256) void sage_gemm_wmma(
    const float* __restrict__ Amean,   // [N, K] neighbor sums
    const float* __restrict__ Aself,   // [N, K]
    const float* __restrict__ deginv,  // [N]
    const float* __restrict__ Wl,      // [K, Cout] row-major
    const float* __restrict__ Wr,      // [K, Cout] row-major
    const float* __restrict__ bias,    // [Cout]
    const float* __restrict__ g,       // [Cout] or nullptr
    const float* __restrict__ be,      // [Cout]
    const float* __restrict__ rm,      // [Cout]
    const float* __restrict__ rv,      // [Cout]
    float* __restrict__ out,           // [N, Cout]
    int K, int Cout, int groups_n, int relu)
{
    const int lane  = threadIdx.x & 31;
    const int wave  = threadIdx.x >> 5;
    const int gtile = blockIdx.x * 8 + wave;
    const int tm    = gtile / groups_n;
    const int gn    = gtile - tm * groups_n;
    if (tm >= TILES_M) return;   // wave-uniform (EXEC stays all-1 for WMMA)

    const int mn    = lane & 15;           // A row-in-tile / B+out col-in-tile
    const int khalf = (lane >> 4) << 1;    // 0 for lanes 0-15, 2 for lanes 16-31
    const int row   = tm * 16 + mn;        // A/deginv row for this lane
    const int col0  = gn * (16 * CT) + mn; // first column handled by this lane

    v8f acc[CT];
#pragma unroll
    for (int j = 0; j < CT; ++j) acc[j] = (v8f){};

    const float dinv = deginv[row];

    // mean @ Wl  (deg_inv folded into the A operand)
    const float* arow = Amean + (size_t)row * K;
    for (int k0 = 0; k0 < K; k0 += 4) {
        const int kk = k0 + khalf;
        const float2 av = *(const float2*)(arow + kk);
        v2f a; a.x = av.x * dinv; a.y = av.y * dinv;
        const float* w0 = Wl + (size_t)kk * Cout + col0;
        const float* w1 = Wl + (size_t)(kk + 1) * Cout + col0;
#pragma unroll
        for (int j = 0; j < CT; ++j) {
            v2f b; b.x = w0[j * 16]; b.y = w1[j * 16];
            acc[j] = __builtin_amdgcn_wmma_f32_16x16x4_f32(false, a, false, b,
                                                           (short)0, acc[j],
                                                           false, false);
        }
    }

    // self @ Wr
    const float* srow = Aself + (size_t)row * K;
    for (int k0 = 0; k0 < K; k0 += 4) {
        const int kk = k0 + khalf;
        const float2 av = *(const float2*)(srow + kk);
        v2f a; a.x = av.x; a.y = av.y;
        const float* w0 = Wr + (size_t)kk * Cout + col0;
        const float* w1 = Wr + (size_t)(kk + 1) * Cout + col0;
#pragma unroll
        for (int j = 0; j < CT; ++j) {
            v2f b; b.x = w0[j * 16]; b.y = w1[j * 16];
            acc[j] = __builtin_amdgcn_wmma_f32_16x16x4_f32(false, a, false, b,
                                                           (short)0, acc[j],
                                                           false, false);
        }
    }

    // Epilogue: bias, folded batch-norm, ReLU
    const int rbase = tm * 16 + ((lane >> 4) << 3);  // lanes 16-31 -> rows +8
#pragma unroll
    for (int j = 0; j < CT; ++j) {
        const int col = col0 + j * 16;
        float scale = 1.0f, shift = 0.0f;
        if (g != nullptr) {
            const float s = g[col] * rsqrtf(rv[col] + 1e-5f);
            scale = s;
            shift = be[col] - rm[col] * s;
        }
        const float bv = bias[col];
#pragma unroll
        for (int r = 0; r < 8; ++r) {
            float val = (acc[j][r] + bv) * scale + shift;
            if (relu) val = fmaxf(val, 0.0f);
            out[(size_t)(rbase + r) * Cout + col] = val;
        }
    }
}

// ---------------------------------------------------------------------------
// Host-side orchestration
// ---------------------------------------------------------------------------
extern "C" void kernel_launch(void* const* d_in, const int* in_sizes, int n_in,
                              void* d_out, int out_size, void* d_ws, size_t ws_size,
                              hipStream_t stream) {
    const float* x   = (const float*)d_in[0];
    const int*   ei  = (const int*)d_in[1];
    const float* W1l = (const float*)d_in[2];
    const float* b1l = (const float*)d_in[3];
    const float* W1r = (const float*)d_in[4];
    const float* g1  = (const float*)d_in[5];
    const float* be1 = (const float*)d_in[6];
    const float* rm1 = (const float*)d_in[7];
    const float* rv1 = (const float*)d_in[8];
    const float* W2l = (const float*)d_in[9];
    const float* b2l = (const float*)d_in[10];
    const float* W2r = (const float*)d_in[11];
    const float* g2  = (const float*)d_in[12];
    const float* be2 = (const float*)d_in[13];
    const float* rm2 = (const float*)d_in[14];
    const float* rv2 = (const float*)d_in[15];
    const float* Wcl = (const float*)d_in[16];
    const float* bcl = (const float*)d_in[17];
    const float* Wcr = (const float*)d_in[18];

    const int* src = ei;
    const int* dst = ei + GNN_E;

    float* ws     = (float*)d_ws;
    float* deginv = ws;                                // N
    float* msg    = ws + GNN_N;                        // N*256 (reused)
    float* h1     = msg + (size_t)GNN_N * 256;         // N*256
    float* h2     = h1 + (size_t)GNN_N * 256;          // N*256

    // ---- degrees ----
    hipMemsetAsync(deginv, 0, (size_t)GNN_N * sizeof(float), stream);
    deg_kernel<<<(GNN_E + 255) / 256, 256, 0, stream>>>(dst, deginv);
    deginv_kernel<<<(GNN_N + 255) / 256, 256, 0, stream>>>(deginv);

    // ---- layer 1: SAGE(128->256) + BN + ReLU ----
    hipMemsetAsync(msg, 0, (size_t)GNN_N * 128 * sizeof(float), stream);
    {
        const unsigned int total = (unsigned int)GNN_E * 32u;  // C4 = 128/4
        scatter_add4<<<(total + 255u) / 256u, 256, 0, stream>>>(x, src, dst, msg, 32, total);
    }
    // 6250 M-tiles * 4 column groups / 8 waves = 3125 blocks (exact)
    sage_gemm_wmma<4><<<(TILES_M * 4) / 8, 256, 0, stream>>>(
        msg, x, deginv, W1l, W1r, b1l, g1, be1, rm1, rv1, h1, 128, 256, 4, 1);

    // ---- layer 2: SAGE(256->256) + BN + ReLU ----
    hipMemsetAsync(msg, 0, (size_t)GNN_N * 256 * sizeof(float), stream);
    {
        const unsigned int total = (unsigned int)GNN_E * 64u;  // C4 = 256/4
        scatter_add4<<<(total + 255u) / 256u, 256, 0, stream>>>(h1, src, dst, msg, 64, total);
    }
    sage_gemm_wmma<4><<<(TILES_M * 4) / 8, 256, 0, stream>>>(
        msg, h1, deginv, W2l, W2r, b2l, g2, be2, rm2, rv2, h2, 256, 256, 4, 1);

    // ---- classifier: SAGE(256->64), no BN / ReLU ----
    hipMemsetAsync(msg, 0, (size_t)GNN_N * 256 * sizeof(float), stream);
    {
        const unsigned int total = (unsigned int)GNN_E * 64u;
        scatter_add4<<<(total + 255u) / 256u, 256, 0, stream>>>(h2, src, dst, msg, 64, total);
    }
    // one column group (Cout=64 = 4 tiles, CT=4): 6250 waves -> 782 blocks (guarded)
    sage_gemm_wmma<4><<<(TILES_M + 7) / 8, 256, 0, stream>>>(
        msg, h2, deginv, Wcl, Wcr, bcl, nullptr, nullptr, nullptr, nullptr,
        (float*)d_out, 256, 64, 1, 0);
}